// Shift_gcn_54760833024239
// MI455X (gfx1250) — compile-verified
//
#include <hip/hip_runtime.h>

// ---------------- problem constants ----------------
#define NB 64
#define CC 64
#define DDIM 64
#define TT 300
#define VV 25
#define L2LEN 325        // T + V
#define CV 1600          // C*V
#define CT 19200         // C*T
#define XS_STRIDE 20800  // CV + CT
#define NTROWS 19200     // N*T
#define YCOLS 1600       // V*D
#define NSTRIDE 480000   // C*T*V (per-n stride of x0)
#define EPSF 1e-5f

typedef __attribute__((ext_vector_type(2))) float v2f;
typedef __attribute__((ext_vector_type(8))) float v8f;

// ---------------- workspace layout (float offsets) ----------------
#define OFF_Y      0ull          // 30,720,000 floats : y (nt,w,d)
#define OFF_XSIN   30720000ull   //  1,331,200 floats : xs input  (n,20800)
#define OFF_XS2    32051200ull   //  1,331,200 floats : xs2 (n,64,325)
#define OFF_XSBAR  33382400ull   //     20,800 floats : mean over n
#define OFF_SCALE  33403200ull   //      1,600 floats : (tanh(sig)+1)[w,c]
#define OFF_STATS  33404800ull   //      6,532 floats : BN statistics
#define OFF_XOFF   33411332ull   //      1,600 ints   : precomputed gather offsets
// stats sub-offsets (floats)
#define ST_SV_SUM 0
#define ST_SV_SQ  1
#define ST_ST_SUM 2
#define ST_ST_SQ  3
#define ST_CH_SUM 4        // 64
#define ST_CH_SQ  68       // 64
#define ST_COL_SUM 132     // 1600
#define ST_COL_SQ  1732    // 1600
#define ST_COL_MEAN 3332   // 1600
#define ST_COL_RSTD 4932   // 1600
#define STATS_FLOATS 6532

__device__ __forceinline__ float hswish(float x) {
    float t = fminf(fmaxf(x + 3.0f, 0.0f), 6.0f);
    return x * t * (1.0f / 6.0f);
}
__device__ __forceinline__ float sigm(float x) { return 1.0f / (1.0f + expf(-x)); }

template <int BS>
__device__ __forceinline__ float blockReduceSum(float v, float* sm) {
    int tid = threadIdx.x;
    __syncthreads();
    sm[tid] = v;
    __syncthreads();
#pragma unroll
    for (int s = BS / 2; s > 0; s >>= 1) {
        if (tid < s) sm[tid] += sm[tid + s];
        __syncthreads();
    }
    return sm[0];
}

// ---- K0: decode shift_in once: xoff[v*64+c] = c2*(T*V) + v2 ----
__global__ void k_prep(const int* __restrict__ shift_in, int* __restrict__ xoff) {
    int id = blockIdx.x * 256 + threadIdx.x;
    if (id >= CV) return;
    int s = shift_in[id];
    xoff[id] = (s & 63) * (TT * VV) + (s >> 6);
}

// ---- K1: xt_pre[n,c,v] = sum_t Wt[t]*x[n,c,t,v] + bt ; accumulate global stats ----
__global__ void k_xt(const float* __restrict__ x0, const int* __restrict__ xoff,
                     const float* __restrict__ Wt, const float* __restrict__ bt,
                     float* __restrict__ xsin, float* __restrict__ stats) {
    __shared__ float sm[256];
    int id = blockIdx.x * 256 + threadIdx.x;  // < 102400 (exact)
    int n = id / CV, rem = id % CV;
    int c = rem / VV, v = rem % VV;
    const float* xp = x0 + (size_t)n * NSTRIDE + xoff[v * CC + c];
    float acc = 0.0f;
    for (int t = 0; t < TT; ++t) acc += Wt[t] * xp[t * VV];
    acc += bt[0];
    xsin[(size_t)n * XS_STRIDE + c * VV + v] = acc;
    float s1 = blockReduceSum<256>(acc, sm);
    float s2 = blockReduceSum<256>(acc * acc, sm);
    if (threadIdx.x == 0) {
        atomicAdd(&stats[ST_ST_SUM], s1);
        atomicAdd(&stats[ST_ST_SQ], s2);
    }
}

// ---- K2: xv_pre[n,c,t] = sum_v Wv[v]*x[n,c,t,v] + bv ; accumulate global stats ----
__global__ void k_xv(const float* __restrict__ x0, const int* __restrict__ xoff,
                     const float* __restrict__ Wv, const float* __restrict__ bv,
                     float* __restrict__ xsin, float* __restrict__ stats) {
    __shared__ float sm[256];
    int id = blockIdx.x * 256 + threadIdx.x;  // < 1,228,800 (exact)
    int n = id / CT, rem = id % CT;
    int c = rem / TT, t = rem % TT;
    const float* xb = x0 + (size_t)n * NSTRIDE + t * VV;
    float acc = 0.0f;
    for (int v = 0; v < VV; ++v) acc += Wv[v] * xb[xoff[v * CC + c]];
    acc += bv[0];
    xsin[(size_t)n * XS_STRIDE + CV + c * TT + t] = acc;
    float s1 = blockReduceSum<256>(acc, sm);
    float s2 = blockReduceSum<256>(acc * acc, sm);
    if (threadIdx.x == 0) {
        atomicAdd(&stats[ST_SV_SUM], s1);
        atomicAdd(&stats[ST_SV_SQ], s2);
    }
}

__global__ void k_fin_scalar(float* stats) {
    float mt = stats[ST_ST_SUM] / (float)(NB * CV);
    float vt = stats[ST_ST_SQ] / (float)(NB * CV) - mt * mt;
    stats[ST_ST_SUM] = mt;
    stats[ST_ST_SQ] = rsqrtf(vt + EPSF);
    float mv = stats[ST_SV_SUM] / (float)(NB * CT);
    float vv = stats[ST_SV_SQ] / (float)(NB * CT) - mv * mv;
    stats[ST_SV_SUM] = mv;
    stats[ST_SV_SQ] = rsqrtf(vv + EPSF);
}

__global__ void k_norm_xsin(float* __restrict__ xsin, const float* __restrict__ stats,
                            const float* gt, const float* bet,
                            const float* gv, const float* bev) {
    int id = blockIdx.x * 256 + threadIdx.x;  // < 1,331,200 (exact)
    int p = id % XS_STRIDE;
    float x = xsin[id];
    float m, r, g, b;
    if (p < CV) { m = stats[ST_ST_SUM]; r = stats[ST_ST_SQ]; g = gt[0]; b = bet[0]; }
    else        { m = stats[ST_SV_SUM]; r = stats[ST_SV_SQ]; g = gv[0]; b = bev[0]; }
    xsin[id] = hswish(g * (x - m) * r + b);
}

// ---- Ws GEMM: xs2[n,o,l] = sum_c Ws[o,c]*xs[n,c,l] + bs[o]; per-channel stats ----
__global__ void k_gemm_ws(const float* __restrict__ xsin, const float* __restrict__ Ws,
                          const float* __restrict__ bs, float* __restrict__ xs2,
                          float* __restrict__ stats) {
    __shared__ float sm[512];
    int n = blockIdx.x >> 6, o = blockIdx.x & 63;
    int l = threadIdx.x;
    float val = 0.0f;
    if (l < L2LEN) {
        const float* xr = xsin + (size_t)n * XS_STRIDE;
        const float* wr = Ws + o * CC;
        for (int c = 0; c < CC; ++c) val += wr[c] * xr[c * L2LEN + l];
        val += bs[o];
        xs2[((size_t)(n * DDIM + o)) * L2LEN + l] = val;
    }
    float contrib = (l < L2LEN) ? val : 0.0f;
    float s1 = blockReduceSum<512>(contrib, sm);
    float s2 = blockReduceSum<512>(contrib * contrib, sm);
    if (threadIdx.x == 0) {
        atomicAdd(&stats[ST_CH_SUM + o], s1);
        atomicAdd(&stats[ST_CH_SQ + o], s2);
    }
}

__global__ void k_fin_ch(float* stats) {
    int o = threadIdx.x;  // 64
    float m = stats[ST_CH_SUM + o] / (float)(NB * L2LEN);
    float v = stats[ST_CH_SQ + o] / (float)(NB * L2LEN) - m * m;
    stats[ST_CH_SUM + o] = m;
    stats[ST_CH_SQ + o] = rsqrtf(v + EPSF);
}

__global__ void k_norm_xs2(float* __restrict__ xs2, const float* __restrict__ stats,
                           const float* gs, const float* bes) {
    int id = blockIdx.x * 256 + threadIdx.x;  // < 1,331,200 (exact)
    int o = (id / L2LEN) & 63;
    float x = xs2[id];
    xs2[id] = hswish(gs[o] * (x - stats[ST_CH_SUM + o]) * stats[ST_CH_SQ + o] + bes[o]);
}

__global__ void k_xsbar(const float* __restrict__ xs2, float* __restrict__ xsbar) {
    int id = blockIdx.x * 256 + threadIdx.x;
    if (id >= DDIM * L2LEN) return;
    int d = id / L2LEN, l = id % L2LEN;
    float a = 0.0f;
    for (int n = 0; n < NB; ++n) a += xs2[((size_t)(n * DDIM + d)) * L2LEN + l];
    xsbar[id] = a * (1.0f / NB);
}

__global__ void k_xtime(const float* __restrict__ xsbar, const float* __restrict__ Wt2,
                        const float* __restrict__ bt2, float* __restrict__ outx) {
    int id = blockIdx.x * 256 + threadIdx.x;  // < 19200 (exact)
    int o = id / TT, l = id % TT;
    float a = 0.0f;
    for (int d = 0; d < DDIM; ++d) a += Wt2[o * DDIM + d] * xsbar[d * L2LEN + l];
    outx[id] = sigm(a + bt2[o]);
}

__global__ void k_scale(const float* __restrict__ xsbar, const float* __restrict__ Wv2,
                        const float* __restrict__ bv2, float* __restrict__ scale) {
    int id = blockIdx.x * 256 + threadIdx.x;
    if (id >= VV * CC) return;
    int w = id / CC, c = id % CC;
    float a = 0.0f;
    for (int d = 0; d < DDIM; ++d) a += Wv2[c * DDIM + d] * xsbar[d * L2LEN + TT + w];
    scale[id] = tanhf(sigm(a + bv2[c])) + 1.0f;
}

// ---- big GEMM via V_WMMA_F32_16X16X4_F32 ----
// y[r, d] = sum_c (x0_gather(r,c)*scale[w,c]) * Lw[c,d] + Lb[d]
// One wave computes a full 16x64 output strip: the gathered A tile is loaded
// once per k-step and reused for all 4 N-tiles (4 accumulator sets, 64 WMMAs).
__global__ void k_gemm_y(const float* __restrict__ x0, const int* __restrict__ xoff,
                         const float* __restrict__ scale, const float* __restrict__ Lw,
                         const float* __restrict__ Lb, float* __restrict__ y) {
    int mt = blockIdx.x * (blockDim.x >> 5) + (threadIdx.x >> 5);  // 0..29999
    int lane = threadIdx.x & 31;
    int mloc = lane & 15;
    int r = mt * 16 + mloc;
    int w = r % VV;
    int ntrow = r / VV;
    int n_ = ntrow / TT, t_ = ntrow % TT;
    int kb = (lane >> 4) << 1;  // K sub-offset: lanes 0-15 -> {0,1}, 16-31 -> {2,3}
    const int* orow = xoff + w * CC;
    const float* scrow = scale + w * CC;
    const float* xb = x0 + (size_t)n_ * NSTRIDE + (size_t)t_ * VV;

    v8f acc0 = {}, acc1 = {}, acc2 = {}, acc3 = {};
#pragma unroll
    for (int kk = 0; kk < 16; ++kk) {
        int k0 = kk * 4 + kb;
        v2f a;
        a[0] = xb[orow[k0]] * scrow[k0];
        a[1] = xb[orow[k0 + 1]] * scrow[k0 + 1];
        const float* l0 = Lw + k0 * DDIM + mloc;  // row k0, col mloc
        const float* l1 = l0 + DDIM;              // row k0+1
        v2f b0; b0[0] = l0[0];  b0[1] = l1[0];
        v2f b1; b1[0] = l0[16]; b1[1] = l1[16];
        v2f b2; b2[0] = l0[32]; b2[1] = l1[32];
        v2f b3; b3[0] = l0[48]; b3[1] = l1[48];
        acc0 = __builtin_amdgcn_wmma_f32_16x16x4_f32(false, a, false, b0, (short)0, acc0, false, false);
        acc1 = __builtin_amdgcn_wmma_f32_16x16x4_f32(false, a, false, b1, (short)0, acc1, false, false);
        acc2 = __builtin_amdgcn_wmma_f32_16x16x4_f32(false, a, false, b2, (short)0, acc2, false, false);
        acc3 = __builtin_amdgcn_wmma_f32_16x16x4_f32(false, a, false, b3, (short)0, acc3, false, false);
    }
    float lb0 = Lb[mloc];
    float lb1 = Lb[mloc + 16];
    float lb2 = Lb[mloc + 32];
    float lb3 = Lb[mloc + 48];
#pragma unroll
    for (int j = 0; j < 8; ++j) {
        int mrow = (lane < 16) ? j : (j + 8);
        float* yr = y + ((size_t)mt * 16 + mrow) * DDIM + mloc;
        yr[0]  = acc0[j] + lb0;
        yr[16] = acc1[j] + lb1;
        yr[32] = acc2[j] + lb2;
        yr[48] = acc3[j] + lb3;
    }
}

// ---- column stats of y over the 19200 nt-rows (per (w,d) column) ----
#define ROWS_PER_BLK 96
__global__ void k_colacc(const float* __restrict__ y, float* __restrict__ stats) {
    int tid = threadIdx.x;  // 256
    float acc[7], accq[7];
#pragma unroll
    for (int i = 0; i < 7; ++i) { acc[i] = 0.0f; accq[i] = 0.0f; }
    int row0 = blockIdx.x * ROWS_PER_BLK;
    for (int rr = 0; rr < ROWS_PER_BLK; ++rr) {
        const float* yr = y + (size_t)(row0 + rr) * YCOLS;
#pragma unroll
        for (int i = 0; i < 7; ++i) {
            int col = tid + i * 256;
            if (col < YCOLS) { float v = yr[col]; acc[i] += v; accq[i] += v * v; }
        }
    }
#pragma unroll
    for (int i = 0; i < 7; ++i) {
        int col = tid + i * 256;
        if (col < YCOLS) {
            atomicAdd(&stats[ST_COL_SUM + col], acc[i]);
            atomicAdd(&stats[ST_COL_SQ + col], accq[i]);
        }
    }
}

__global__ void k_colfin(float* stats) {
    int c = blockIdx.x * 256 + threadIdx.x;
    if (c >= YCOLS) return;
    float m = stats[ST_COL_SUM + c] / (float)NTROWS;
    float v = stats[ST_COL_SQ + c] / (float)NTROWS - m * m;
    stats[ST_COL_MEAN + c] = m;
    stats[ST_COL_RSTD + c] = rsqrtf(v + EPSF);
}

// ---- final: shift_out permute + column BN + residual + ReLU ----
__global__ void k_final(const float* __restrict__ y, const float* __restrict__ x0,
                        const int* __restrict__ shift_out, const float* __restrict__ gbn,
                        const float* __restrict__ bbn, const float* __restrict__ stats,
                        float* __restrict__ out) {
    size_t id = (size_t)blockIdx.x * 256 + threadIdx.x;  // < 30,720,000 (exact)
    int v = (int)(id % VV);
    size_t q = id / VV;
    int t = (int)(q % TT); q /= TT;
    int dd = (int)(q % DDIM);
    int n = (int)(q / DDIM);
    int k = v * DDIM + dd;
    int c0 = shift_out[k];
    size_t nt = (size_t)n * TT + t;
    float yv = y[nt * YCOLS + c0];
    float val = gbn[k] * (yv - stats[ST_COL_MEAN + c0]) * stats[ST_COL_RSTD + c0] + bbn[k] + x0[id];
    out[id] = fmaxf(val, 0.0f);
}

extern "C" void kernel_launch(void* const* d_in, const int* in_sizes, int n_in,
                              void* d_out, int out_size, void* d_ws, size_t ws_size,
                              hipStream_t stream) {
    const float* x0      = (const float*)d_in[0];
    const int*   shift_in  = (const int*)d_in[1];
    const int*   shift_out = (const int*)d_in[2];
    const float* Wv  = (const float*)d_in[3];
    const float* bv  = (const float*)d_in[4];
    const float* gv  = (const float*)d_in[5];
    const float* bev = (const float*)d_in[6];
    const float* Wt  = (const float*)d_in[7];
    const float* bt  = (const float*)d_in[8];
    const float* gt  = (const float*)d_in[9];
    const float* bet = (const float*)d_in[10];
    const float* Ws  = (const float*)d_in[11];
    const float* bs  = (const float*)d_in[12];
    const float* gs  = (const float*)d_in[13];
    const float* bes = (const float*)d_in[14];
    const float* Wv2 = (const float*)d_in[15];
    const float* bv2 = (const float*)d_in[16];
    const float* Wt2 = (const float*)d_in[17];
    const float* bt2 = (const float*)d_in[18];
    const float* Lw  = (const float*)d_in[19];
    const float* Lb  = (const float*)d_in[20];
    const float* gbn = (const float*)d_in[21];
    const float* bbn = (const float*)d_in[22];

    float* W     = (float*)d_ws;
    float* y     = W + OFF_Y;
    float* xsin  = W + OFF_XSIN;
    float* xs2   = W + OFF_XS2;
    float* xsbar = W + OFF_XSBAR;
    float* scale = W + OFF_SCALE;
    float* stats = W + OFF_STATS;
    int*   xoff  = (int*)(W + OFF_XOFF);
    float* out   = (float*)d_out;
    float* out_xtime = out + 30720000;  // second tuple output (1,64,300)

    hipMemsetAsync(stats, 0, STATS_FLOATS * sizeof(float), stream);

    k_prep      <<<7,      256, 0, stream>>>(shift_in, xoff);
    k_xt        <<<400,    256, 0, stream>>>(x0, xoff, Wt, bt, xsin, stats);
    k_xv        <<<4800,   256, 0, stream>>>(x0, xoff, Wv, bv, xsin, stats);
    k_fin_scalar<<<1,        1, 0, stream>>>(stats);
    k_norm_xsin <<<5200,   256, 0, stream>>>(xsin, stats, gt, bet, gv, bev);
    k_gemm_ws   <<<4096,   512, 0, stream>>>(xsin, Ws, bs, xs2, stats);
    k_fin_ch    <<<1,       64, 0, stream>>>(stats);
    k_norm_xs2  <<<5200,   256, 0, stream>>>(xs2, stats, gs, bes);
    k_xsbar     <<<82,     256, 0, stream>>>(xs2, xsbar);
    k_xtime     <<<75,     256, 0, stream>>>(xsbar, Wt2, bt2, out_xtime);
    k_scale     <<<7,      256, 0, stream>>>(xsbar, Wv2, bv2, scale);
    k_gemm_y    <<<3750,   256, 0, stream>>>(x0, xoff, scale, Lw, Lb, y);
    k_colacc    <<<200,    256, 0, stream>>>(y, stats);
    k_colfin    <<<7,      256, 0, stream>>>(stats);
    k_final     <<<120000, 256, 0, stream>>>(y, x0, shift_out, gbn, bbn, stats, out);
}